// GraphAttentionNetwork_59914793779546
// MI455X (gfx1250) — compile-verified
//
#include <hip/hip_runtime.h>
#include <hip/hip_bf16.h>
#include <math.h>

// ---------------------------------------------------------------------------
// Problem constants (match reference)
// ---------------------------------------------------------------------------
#define NNODES 100000
#define NEDGES 800000
#define NGRAPH 2000
#define NHEADS 4
#define NEG_SLOPE 0.2f
#define BN_EPS 1e-5f

static_assert(NNODES % 16 == 0, "row-guard elimination requires N % 16 == 0");

typedef __attribute__((ext_vector_type(2)))  float v2f;
typedef __attribute__((ext_vector_type(8)))  float v8f;

// ---------------------------------------------------------------------------
// Utility fills (graph-capture-safe initialization of workspace accumulators)
// ---------------------------------------------------------------------------
__global__ void fill_f32_kernel(float* __restrict__ p, float v, int n) {
    int i = blockIdx.x * blockDim.x + threadIdx.x;
    if (i < n) p[i] = v;
}

// Pad layer-0 features [N,9] -> [N,16] (zero-filled).
__global__ void pad_x0_kernel(const float* __restrict__ x9, float* __restrict__ x16, int n) {
    int i = blockIdx.x * blockDim.x + threadIdx.x;     // over N*16
    if (i >= n * 16) return;
    const int r = i >> 4;
    const int c = i & 15;
    x16[i] = (c < 9) ? x9[r * 9 + c] : 0.0f;
}

// Pad layer-0 weight [9,256] -> [16,256] (zero-filled rows 9..15).
__global__ void pad_w0_kernel(const float* __restrict__ w9, float* __restrict__ w16) {
    int i = blockIdx.x * blockDim.x + threadIdx.x;     // over 16*256
    if (i >= 16 * 256) return;
    const int r = i >> 8;          // /256
    w16[i] = (r < 9) ? w9[i] : 0.0f;
}

// ---------------------------------------------------------------------------
// Dense feature transform: H = X @ W   (X: [N,K], W: [K,HC], H: [N,HC])
// Compile-time K/HC: fully unrolled, branch-free, one wave per 16-row block,
// A fragments held in registers and reused across all HC/16 column tiles.
//
// ISA lane layouts (CDNA5 ISA, f32 16x16x4):
//   A 16x4 :  lane&15 = row;  a.x = A[row][2*hi], a.y = A[row][2*hi+1]
//   B 4x16 :  lane&15 = col;  b.x = B[2*hi][col], b.y = B[2*hi+1][col]
//   D 16x16:  acc[j] -> D[j + 8*hi][col]
// ---------------------------------------------------------------------------
template<int K, int HC>
__global__ __launch_bounds__(32)
void gat_gemm_wmma_f32(const float* __restrict__ X, const float* __restrict__ W,
                       float* __restrict__ H) {
    const int m0   = blockIdx.x * 16;
    const int lane = threadIdx.x;          // 0..31, full wave (EXEC all ones)
    const int hi   = lane >> 4;            // 0 or 1
    const int lo   = lane & 15;
    const int row  = m0 + lo;

    // Preload this wave's A fragments (K even, ka even -> aligned b64 loads)
    v2f afrag[K / 4];
    #pragma unroll
    for (int t = 0; t < K / 4; ++t) {
        const int ka = 4 * t + 2 * hi;
        afrag[t] = *(const v2f*)(X + (size_t)row * K + ka);
    }

    #pragma unroll
    for (int n0 = 0; n0 < HC; n0 += 16) {
        const int col = n0 + lo;
        v8f acc = {};
        #pragma unroll
        for (int t = 0; t < K / 4; ++t) {
            const int ka = 4 * t + 2 * hi;
            v2f b;
            b.x = W[(size_t)ka       * HC + col];
            b.y = W[(size_t)(ka + 1) * HC + col];
#if __has_builtin(__builtin_amdgcn_wmma_f32_16x16x4_f32)
            acc = __builtin_amdgcn_wmma_f32_16x16x4_f32(
                /*neg_a=*/false, afrag[t], /*neg_b=*/false, b,
                /*c_mod=*/(short)0, acc, /*reuse_a=*/false, /*reuse_b=*/false);
#else
            #pragma unroll
            for (int j = 0; j < 8; ++j) {
                const int r = m0 + j + 8 * hi;
                float s = 0.f;
                for (int kk = 0; kk < 4; ++kk)
                    s += X[(size_t)r * K + 4 * t + kk] * W[(size_t)(4 * t + kk) * HC + col];
                acc[j] += s;
            }
#endif
        }
        #pragma unroll
        for (int j = 0; j < 8; ++j)
            H[(size_t)(m0 + j + 8 * hi) * HC + col] = acc[j];
    }
}

// ---------------------------------------------------------------------------
// Per-node attention coefficients: a_src[n,h] = <h[n,h,:], att_src[h,:]>
// ---------------------------------------------------------------------------
__global__ void gat_att_coef(const float* __restrict__ H,
                             const float* __restrict__ attS,
                             const float* __restrict__ attD,
                             float* __restrict__ aS, float* __restrict__ aD,
                             int nrows, int cout) {
    int i = blockIdx.x * blockDim.x + threadIdx.x;   // over N*HEADS
    if (i >= nrows * NHEADS) return;
    const int n = i / NHEADS;
    const int h = i - n * NHEADS;
    const int HC = NHEADS * cout;
    const float4* hp = (const float4*)(H + (size_t)n * HC + h * cout);
    const float4* sp = (const float4*)(attS + h * cout);
    const float4* dp = (const float4*)(attD + h * cout);
    float s = 0.f, d = 0.f;
    for (int c4 = 0; c4 < (cout >> 2); ++c4) {       // cout in {64,32,16,8}
        float4 v = hp[c4], a = sp[c4], b = dp[c4];
        s += v.x * a.x + v.y * a.y + v.z * a.z + v.w * a.w;
        d += v.x * b.x + v.y * b.y + v.z * b.z + v.w * b.w;
    }
    aS[i] = s;
    aD[i] = d;
}

// ---------------------------------------------------------------------------
// Ordered-float atomic max (init to -inf; correct for mixed signs)
// ---------------------------------------------------------------------------
__device__ __forceinline__ void atomicMaxF(float* addr, float val) {
    if (val >= 0.0f) atomicMax((int*)addr, __float_as_int(val));
    else             atomicMin((unsigned int*)addr, __float_as_uint(val));
}

__device__ __forceinline__ float leaky_relu(float x) {
    return x >= 0.0f ? x : NEG_SLOPE * x;
}

// Edge pass 1: segment max of e = leakyrelu(a_src[s]+a_dst[d]) per (dst, head)
__global__ void gat_edge_max(const int* __restrict__ src, const int* __restrict__ dst,
                             const float* __restrict__ aS, const float* __restrict__ aD,
                             float* __restrict__ emax, int nedges, int nrows) {
    int i = blockIdx.x * blockDim.x + threadIdx.x;   // over E + N (self-loops)
    if (i >= nedges + nrows) return;
    const int s = (i < nedges) ? src[i] : (i - nedges);
    const int d = (i < nedges) ? dst[i] : (i - nedges);
    const float4 vs = *(const float4*)(aS + s * NHEADS);
    const float4 vd = *(const float4*)(aD + d * NHEADS);
    atomicMaxF(&emax[d * NHEADS + 0], leaky_relu(vs.x + vd.x));
    atomicMaxF(&emax[d * NHEADS + 1], leaky_relu(vs.y + vd.y));
    atomicMaxF(&emax[d * NHEADS + 2], leaky_relu(vs.z + vd.z));
    atomicMaxF(&emax[d * NHEADS + 3], leaky_relu(vs.w + vd.w));
}

// Edge pass 2: denom[d,h] += exp(e - emax[d,h])
__global__ void gat_edge_sum(const int* __restrict__ src, const int* __restrict__ dst,
                             const float* __restrict__ aS, const float* __restrict__ aD,
                             const float* __restrict__ emax, float* __restrict__ denom,
                             int nedges, int nrows) {
    int i = blockIdx.x * blockDim.x + threadIdx.x;
    if (i >= nedges + nrows) return;
    const int s = (i < nedges) ? src[i] : (i - nedges);
    const int d = (i < nedges) ? dst[i] : (i - nedges);
    const float4 vs = *(const float4*)(aS + s * NHEADS);
    const float4 vd = *(const float4*)(aD + d * NHEADS);
    const float4 vm = *(const float4*)(emax + d * NHEADS);
    atomicAdd(&denom[d * NHEADS + 0], __expf(leaky_relu(vs.x + vd.x) - vm.x));
    atomicAdd(&denom[d * NHEADS + 1], __expf(leaky_relu(vs.y + vd.y) - vm.y));
    atomicAdd(&denom[d * NHEADS + 2], __expf(leaky_relu(vs.z + vd.z) - vm.z));
    atomicAdd(&denom[d * NHEADS + 3], __expf(leaky_relu(vs.w + vd.w) - vm.w));
}

// Edge pass 3: agg[d,h,:] += alpha * h[s,h,:]   (alpha recomputed, no edge storage)
__global__ void gat_edge_aggregate(const int* __restrict__ src, const int* __restrict__ dst,
                                   const float* __restrict__ aS, const float* __restrict__ aD,
                                   const float* __restrict__ emax, const float* __restrict__ denom,
                                   const float* __restrict__ H, float* __restrict__ agg,
                                   int nedges, int nrows, int cout) {
    int i = blockIdx.x * blockDim.x + threadIdx.x;   // over (E+N)*HEADS
    if (i >= (nedges + nrows) * NHEADS) return;
    const int e = i / NHEADS;
    const int h = i - e * NHEADS;
    const int s = (e < nedges) ? src[e] : (e - nedges);
    const int d = (e < nedges) ? dst[e] : (e - nedges);

    const float lg    = leaky_relu(aS[s * NHEADS + h] + aD[d * NHEADS + h]);
    const float alpha = __expf(lg - emax[d * NHEADS + h]) /
                        (denom[d * NHEADS + h] + 1e-16f);

    const int HC = NHEADS * cout;
    const float4* hp = (const float4*)(H + (size_t)s * HC + h * cout);
    float* ap = agg + (size_t)d * HC + h * cout;
    for (int c4 = 0; c4 < (cout >> 2); ++c4) {       // cout in {64,32,16,8}
        float4 v = hp[c4];
        atomicAdd(&ap[c4 * 4 + 0], alpha * v.x);
        atomicAdd(&ap[c4 * 4 + 1], alpha * v.y);
        atomicAdd(&ap[c4 * 4 + 2], alpha * v.z);
        atomicAdd(&ap[c4 * 4 + 3], alpha * v.w);
    }
}

// Finalize: head-mean + bias, eval-BatchNorm, ELU -> next layer input
__global__ void gat_finalize(const float* __restrict__ agg,
                             const float* __restrict__ bias,  const float* __restrict__ gamma,
                             const float* __restrict__ beta,  const float* __restrict__ mean,
                             const float* __restrict__ var,
                             float* __restrict__ xout, int nrows, int cout) {
    int i = blockIdx.x * blockDim.x + threadIdx.x;   // over N*cout
    if (i >= nrows * cout) return;
    const int n = i / cout;
    const int c = i - n * cout;
    const int HC = NHEADS * cout;
    float v = 0.f;
    #pragma unroll
    for (int h = 0; h < NHEADS; ++h) v += agg[(size_t)n * HC + h * cout + c];
    v = v * (1.0f / NHEADS) + bias[c];
    v = (v - mean[c]) * rsqrtf(var[c] + BN_EPS) * gamma[c] + beta[c];
    v = (v > 0.0f) ? v : (__expf(v) - 1.0f);         // ELU(alpha=1)
    xout[i] = v;
}

// ---------------------------------------------------------------------------
// Global mean pool + linear readout
// ---------------------------------------------------------------------------
__global__ void pool_accum(const float* __restrict__ x, const int* __restrict__ batch,
                           float* __restrict__ pooled, float* __restrict__ counts,
                           int nrows, int c) {
    int i = blockIdx.x * blockDim.x + threadIdx.x;
    if (i >= nrows) return;
    const int b = batch[i];
    for (int j = 0; j < c; ++j) atomicAdd(&pooled[b * c + j], x[(size_t)i * c + j]);
    atomicAdd(&counts[b], 1.0f);
}

__global__ void pool_readout(const float* __restrict__ pooled, const float* __restrict__ counts,
                             const float* __restrict__ Wout, const float* __restrict__ bout,
                             float* __restrict__ out, int g, int c) {
    int i = blockIdx.x * blockDim.x + threadIdx.x;
    if (i >= g) return;
    const float inv = 1.0f / fmaxf(counts[i], 1.0f);
    float acc = bout[0];
    for (int j = 0; j < c; ++j) acc += pooled[i * c + j] * inv * Wout[j];
    out[i] = acc;
}

// ---------------------------------------------------------------------------
// Host-side launcher
// Input order (setup_inputs dict, params list flattened per-layer):
//   0: x [N,9]
//   1 + 8*L + {0..7}: W, att_src, att_dst, bias, gamma, beta, mean, var
//   33: Wout [8,1]   34: bout [1]   35: src [E]   36: dst [E]   37: batch [N]
// ---------------------------------------------------------------------------
extern "C" void kernel_launch(void* const* d_in, const int* in_sizes, int n_in,
                              void* d_out, int out_size, void* d_ws, size_t ws_size,
                              hipStream_t stream) {
    (void)in_sizes; (void)n_in; (void)ws_size; (void)out_size;
    static const int COUT[4] = {64, 32, 16, 8};

    const float* Wout  = (const float*)d_in[33];
    const float* bout  = (const float*)d_in[34];
    const int*   src   = (const int*)d_in[35];
    const int*   dst   = (const int*)d_in[36];
    const int*   batch = (const int*)d_in[37];
    float*       out   = (float*)d_out;

    // Workspace carve-up (fp32); all offsets multiples of 16 floats (64B aligned)
    float* ws     = (float*)d_ws;
    float* xbuf   = ws;                                  // N*64
    float* hfeat  = xbuf  + (size_t)NNODES * 64;         // N*256
    float* agg    = hfeat + (size_t)NNODES * 256;        // N*256
    float* xpad   = agg   + (size_t)NNODES * 256;        // N*16 (padded layer-0 input)
    float* w0pad  = xpad  + (size_t)NNODES * 16;         // 16*256 (padded layer-0 weight)
    float* aS     = w0pad + (size_t)16 * 256;            // N*4
    float* aD     = aS    + (size_t)NNODES * NHEADS;     // N*4
    float* emax   = aD    + (size_t)NNODES * NHEADS;     // N*4
    float* denom  = emax  + (size_t)NNODES * NHEADS;     // N*4
    float* pooled = denom + (size_t)NNODES * NHEADS;     // G*8
    float* counts = pooled + (size_t)NGRAPH * 8;         // G

    const int TB = 256;
    const int nEdgeTot = NEDGES + NNODES;                // edges + self-loops
    const int MBLK = NNODES / 16;                        // 6250 row blocks

    // Zero-pad layer-0 input and weight once per call (graph-capture safe)
    pad_x0_kernel<<<(NNODES * 16 + TB - 1) / TB, TB, 0, stream>>>(
        (const float*)d_in[0], xpad, NNODES);
    pad_w0_kernel<<<(16 * 256 + TB - 1) / TB, TB, 0, stream>>>(
        (const float*)d_in[1], w0pad);

    for (int L = 0; L < 4; ++L) {
        const int cout = COUT[L];
        const int HC   = NHEADS * cout;
        const float* W     = (const float*)d_in[1 + 8 * L + 0];
        const float* attS  = (const float*)d_in[1 + 8 * L + 1];
        const float* attD  = (const float*)d_in[1 + 8 * L + 2];
        const float* bias  = (const float*)d_in[1 + 8 * L + 3];
        const float* gamma = (const float*)d_in[1 + 8 * L + 4];
        const float* beta  = (const float*)d_in[1 + 8 * L + 5];
        const float* mean  = (const float*)d_in[1 + 8 * L + 6];
        const float* var   = (const float*)d_in[1 + 8 * L + 7];

        // Init accumulators for this layer
        {
            int nAgg = NNODES * HC;
            fill_f32_kernel<<<(nAgg + TB - 1) / TB, TB, 0, stream>>>(agg, 0.0f, nAgg);
            int nNH = NNODES * NHEADS;
            fill_f32_kernel<<<(nNH + TB - 1) / TB, TB, 0, stream>>>(emax, -__builtin_inff(), nNH);
            fill_f32_kernel<<<(nNH + TB - 1) / TB, TB, 0, stream>>>(denom, 0.0f, nNH);
        }

        // 1) h = x @ W on the WMMA pipe (exact fp32; branch-free, unrolled)
        switch (L) {
            case 0: gat_gemm_wmma_f32<16, 256><<<MBLK, 32, 0, stream>>>(xpad, w0pad, hfeat); break;
            case 1: gat_gemm_wmma_f32<64, 128><<<MBLK, 32, 0, stream>>>(xbuf, W, hfeat); break;
            case 2: gat_gemm_wmma_f32<32,  64><<<MBLK, 32, 0, stream>>>(xbuf, W, hfeat); break;
            case 3: gat_gemm_wmma_f32<16,  32><<<MBLK, 32, 0, stream>>>(xbuf, W, hfeat); break;
        }

        // 2) attention coefficients
        {
            int n = NNODES * NHEADS;
            gat_att_coef<<<(n + TB - 1) / TB, TB, 0, stream>>>(hfeat, attS, attD, aS, aD,
                                                               NNODES, cout);
        }
        // 3) segment softmax over destination nodes
        gat_edge_max<<<(nEdgeTot + TB - 1) / TB, TB, 0, stream>>>(src, dst, aS, aD, emax,
                                                                  NEDGES, NNODES);
        gat_edge_sum<<<(nEdgeTot + TB - 1) / TB, TB, 0, stream>>>(src, dst, aS, aD, emax,
                                                                  denom, NEDGES, NNODES);
        // 4) weighted scatter aggregation
        {
            int n = nEdgeTot * NHEADS;
            gat_edge_aggregate<<<(n + TB - 1) / TB, TB, 0, stream>>>(
                src, dst, aS, aD, emax, denom, hfeat, agg, NEDGES, NNODES, cout);
        }
        // 5) head-mean + bias + BN + ELU -> xbuf (next layer input)
        {
            int n = NNODES * cout;
            gat_finalize<<<(n + TB - 1) / TB, TB, 0, stream>>>(agg, bias, gamma, beta,
                                                               mean, var, xbuf, NNODES, cout);
        }
    }

    // Global mean pool + linear readout
    {
        int nP = NGRAPH * 8;
        fill_f32_kernel<<<(nP + TB - 1) / TB, TB, 0, stream>>>(pooled, 0.0f, nP);
        fill_f32_kernel<<<(NGRAPH + TB - 1) / TB, TB, 0, stream>>>(counts, 0.0f, NGRAPH);
        pool_accum<<<(NNODES + TB - 1) / TB, TB, 0, stream>>>(xbuf, batch, pooled, counts,
                                                              NNODES, 8);
        pool_readout<<<(NGRAPH + TB - 1) / TB, TB, 0, stream>>>(pooled, counts, Wout, bout,
                                                                out, NGRAPH, 8);
    }
}